// StructuralAttentionLayer_35476429865414
// MI455X (gfx1250) — compile-verified
//
#include <hip/hip_runtime.h>
#include <math.h>

typedef float v2f __attribute__((ext_vector_type(2)));
typedef float v8f __attribute__((ext_vector_type(8)));

#define HCX 256   // H*C
#define HH  8
#define CC  32

// ---------------------------------------------------------------------------
// GEMM: Cout[M,256] = A[M,256] @ B[256,256] using V_WMMA_F32_16X16X4_F32.
// One wave32 per 16x16 output tile, 8 waves (256 threads) per block.
// A frag (16x4, 2 VGPR): lane<16 -> row=lane,   K = k+0 (v0), k+1 (v1)
//                        lane>=16-> row=lane-16,K = k+2 (v0), k+3 (v1)
// B frag (4x16, 2 VGPR): col = lane%16, K = k + 2*(lane>=16) + vgpr
// C/D (16x16, 8 VGPR):   col = lane%16, row = vgpr + 8*(lane>=16)
// ---------------------------------------------------------------------------
__global__ __launch_bounds__(256) void gat_gemm256_wmma(
    const float* __restrict__ A, const float* __restrict__ B,
    float* __restrict__ Cout, int M)
{
    const int lane = threadIdx.x & 31;
    const int wave = threadIdx.x >> 5;
    const int tile = blockIdx.x * 8 + wave;
    const int tm   = tile >> 4;     // 256/16 = 16 column tiles
    const int tn   = tile & 15;
    if ((tm << 4) >= M) return;     // uniform per wave; no barriers used

    const int half = lane >> 4;     // 0 or 1
    const int l16  = lane & 15;
    const long arow = (long)(tm * 16 + l16);
    const int  bcol = tn * 16 + l16;

    const float* Aptr = A + arow * HCX + 2 * half;      // 8B aligned (k even)
    const float* Bptr = B + (size_t)(2 * half) * HCX + bcol;

    v8f acc = {};
    #pragma unroll 4
    for (int k = 0; k < HCX; k += 4) {
        v2f a = *(const v2f*)(Aptr + k);                // A[row][k+2h], A[row][k+2h+1]
        v2f b;
        b.x = Bptr[(size_t)k * HCX];                    // B[k+2h  ][col]
        b.y = Bptr[(size_t)k * HCX + HCX];              // B[k+2h+1][col]
        acc = __builtin_amdgcn_wmma_f32_16x16x4_f32(
            /*neg_a=*/false, a, /*neg_b=*/false, b,
            /*c_mod=*/(short)0, acc, /*reuse_a=*/false, /*reuse_b=*/false);
    }

    float* crow = Cout + (long)(tm * 16 + 8 * half) * HCX + bcol;
    #pragma unroll
    for (int v = 0; v < 8; ++v)
        crow[(long)v * HCX] = acc[v];
}

// ---------------------------------------------------------------------------
// alpha_l[n,h] = sum_c x[n,h,c]*att_l[h,c]; alpha_r likewise.
// ---------------------------------------------------------------------------
__global__ void gat_alpha(const float* __restrict__ x,
                          const float* __restrict__ att_l,
                          const float* __restrict__ att_r,
                          float* __restrict__ alpha_l,
                          float* __restrict__ alpha_r, int N)
{
    int t = blockIdx.x * blockDim.x + threadIdx.x;
    if (t >= N * HH) return;
    int n = t >> 3, h = t & 7;
    const float4* xr = (const float4*)(x + (size_t)n * HCX + h * CC);
    const float4* lv = (const float4*)(att_l + h * CC);
    const float4* rv = (const float4*)(att_r + h * CC);
    float sl = 0.f, sr = 0.f;
    #pragma unroll
    for (int i = 0; i < 8; ++i) {
        float4 xv = xr[i], lw = lv[i], rw = rv[i];
        sl += xv.x * lw.x + xv.y * lw.y + xv.z * lw.z + xv.w * lw.w;
        sr += xv.x * rw.x + xv.y * rw.y + xv.z * rw.z + xv.w * rw.w;
    }
    alpha_l[t] = sl;
    alpha_r[t] = sr;
}

// ---------------------------------------------------------------------------
__global__ void gat_init(float* __restrict__ agg, float* __restrict__ amax,
                         float* __restrict__ denom, int N)
{
    int t = blockIdx.x * blockDim.x + threadIdx.x;
    if (t < N * HCX) agg[t] = 0.f;
    if (t < N * HH) { amax[t] = -INFINITY; denom[t] = 0.f; }
}

__device__ __forceinline__ void atomicMaxFloat(float* addr, float v) {
    if (v >= 0.f) atomicMax((int*)addr, __float_as_int(v));
    else          atomicMin((unsigned int*)addr, __float_as_uint(v));
}

__device__ __forceinline__ float edge_score(const float* __restrict__ ew,
                                            const float* __restrict__ al,
                                            const float* __restrict__ ar,
                                            int e, int s, int d, int h)
{
    float a = ew[e] * (al[(size_t)s * HH + h] + ar[(size_t)d * HH + h]);
    return a > 0.f ? a : 0.2f * a;   // leaky_relu(0.2)
}

// Pass 1: segment max over destination. one thread per (edge, head)
__global__ void gat_edge_max(const int* __restrict__ ei, const float* __restrict__ ew,
                             const float* __restrict__ al, const float* __restrict__ ar,
                             float* __restrict__ amax, int E)
{
    int t = blockIdx.x * blockDim.x + threadIdx.x;
    if (t >= E * HH) return;
    int e = t >> 3, h = t & 7;
    int s = ei[e], d = ei[E + e];
    float a = edge_score(ew, al, ar, e, s, d, h);
    atomicMaxFloat(&amax[(size_t)d * HH + h], a);
}

// Pass 2: segment sum of exp(alpha - amax)
__global__ void gat_edge_denom(const int* __restrict__ ei, const float* __restrict__ ew,
                               const float* __restrict__ al, const float* __restrict__ ar,
                               const float* __restrict__ amax, float* __restrict__ denom, int E)
{
    int t = blockIdx.x * blockDim.x + threadIdx.x;
    if (t >= E * HH) return;
    int e = t >> 3, h = t & 7;
    int s = ei[e], d = ei[E + e];
    float a  = edge_score(ew, al, ar, e, s, d, h);
    float ex = expf(a - amax[(size_t)d * HH + h]);
    atomicAdd(&denom[(size_t)d * HH + h], ex);
}

// Pass 3: weighted message scatter. One wave32 per edge; lane owns 8 channels
// of one head (h = lane/4, c0 = (lane%4)*8). x/agg are L2-resident (51MB each).
__global__ __launch_bounds__(256) void gat_edge_aggregate(
    const int* __restrict__ ei, const float* __restrict__ ew,
    const float* __restrict__ al, const float* __restrict__ ar,
    const float* __restrict__ amax, const float* __restrict__ denom,
    const float* __restrict__ x, float* __restrict__ agg, int E)
{
    const int lane = threadIdx.x & 31;
    const int wid  = threadIdx.x >> 5;
    const int e    = blockIdx.x * 8 + wid;
    if (e >= E) return;

    const int s = ei[e], d = ei[E + e];
    const int h  = lane >> 2;
    const int c0 = (lane & 3) * 8;

    float a     = edge_score(ew, al, ar, e, s, d, h);
    float ex    = expf(a - amax[(size_t)d * HH + h]);
    float coeff = ex / (denom[(size_t)d * HH + h] + 1e-16f);

    const float* xs = x   + (size_t)s * HCX + h * CC + c0;
    float*       ag = agg + (size_t)d * HCX + h * CC + c0;
    float4 x0 = *(const float4*)(xs);
    float4 x1 = *(const float4*)(xs + 4);
    atomicAdd(ag + 0, x0.x * coeff);
    atomicAdd(ag + 1, x0.y * coeff);
    atomicAdd(ag + 2, x0.z * coeff);
    atomicAdd(ag + 3, x0.w * coeff);
    atomicAdd(ag + 4, x1.x * coeff);
    atomicAdd(ag + 5, x1.y * coeff);
    atomicAdd(ag + 6, x1.z * coeff);
    atomicAdd(ag + 7, x1.w * coeff);
}

// out (already holds feat@W_res) += elu(agg)
__global__ void gat_finalize(const float* __restrict__ agg, float* __restrict__ out, int N)
{
    int t = blockIdx.x * blockDim.x + threadIdx.x;
    if (t >= N * HCX) return;
    float a = agg[t];
    out[t] += (a > 0.f) ? a : (expf(a) - 1.0f);
}

// ---------------------------------------------------------------------------
extern "C" void kernel_launch(void* const* d_in, const int* in_sizes, int n_in,
                              void* d_out, int out_size, void* d_ws, size_t ws_size,
                              hipStream_t stream)
{
    (void)n_in; (void)out_size; (void)ws_size;
    const float* feat = (const float*)d_in[0];
    const float* ew   = (const float*)d_in[1];
    const float* Wlin = (const float*)d_in[2];
    const float* attl = (const float*)d_in[3];
    const float* attr = (const float*)d_in[4];
    const float* Wres = (const float*)d_in[5];
    const int*   ei   = (const int*)d_in[6];
    float* out = (float*)d_out;

    const int N = in_sizes[0] / HCX;
    const int E = in_sizes[1];

    char* ws = (char*)d_ws;
    size_t off = 0;
    auto carve = [&](size_t bytes) -> void* {
        void* p = ws + off;
        off += (bytes + 255) & ~(size_t)255;
        return p;
    };
    float* x       = (float*)carve((size_t)N * HCX * sizeof(float));
    float* agg     = (float*)carve((size_t)N * HCX * sizeof(float));
    float* alpha_l = (float*)carve((size_t)N * HH * sizeof(float));
    float* alpha_r = (float*)carve((size_t)N * HH * sizeof(float));
    float* amax    = (float*)carve((size_t)N * HH * sizeof(float));
    float* denom   = (float*)carve((size_t)N * HH * sizeof(float));

    // init scratch (agg=0, amax=-inf, denom=0)
    {
        int total = N * HCX;
        gat_init<<<(total + 255) / 256, 256, 0, stream>>>(agg, amax, denom, N);
    }
    // GEMMs via WMMA: x = feat@W_lin ; out = feat@W_res
    {
        int mt = (N + 15) / 16;
        int tiles = mt * 16;
        int blocks = (tiles + 7) / 8;
        gat_gemm256_wmma<<<blocks, 256, 0, stream>>>(feat, Wlin, x, N);
        gat_gemm256_wmma<<<blocks, 256, 0, stream>>>(feat, Wres, out, N);
    }
    // per-node attention logits
    {
        int total = N * HH;
        gat_alpha<<<(total + 255) / 256, 256, 0, stream>>>(x, attl, attr, alpha_l, alpha_r, N);
    }
    // segment softmax (max, denom) + message scatter
    {
        int totalEH = E * HH;
        gat_edge_max  <<<(totalEH + 255) / 256, 256, 0, stream>>>(ei, ew, alpha_l, alpha_r, amax, E);
        gat_edge_denom<<<(totalEH + 255) / 256, 256, 0, stream>>>(ei, ew, alpha_l, alpha_r, amax, denom, E);
        int blocks = (E + 7) / 8;   // 8 edges (waves) per 256-thread block
        gat_edge_aggregate<<<blocks, 256, 0, stream>>>(ei, ew, alpha_l, alpha_r, amax, denom, x, agg, E);
    }
    // out += elu(agg)
    {
        int total = N * HCX;
        gat_finalize<<<(total + 255) / 256, 256, 0, stream>>>(agg, out, N);
    }
}